// SubSamplingLayer_53626961657924
// MI455X (gfx1250) — compile-verified
//
#include <hip/hip_runtime.h>
#include <hip/hip_bf16.h>

typedef float v2f __attribute__((ext_vector_type(2)));
typedef float v8f __attribute__((ext_vector_type(8)));

#define RES   320
#define HW    (RES * RES)          // 102400
#define NTRAJ (RES * RES)          // 102400 trajectory points
#define NB    64

// ---------------------------------------------------------------------------
// 1) Build centered orthonormal inverse-DFT matrix W (symmetric, complex).
//    W[m,k] = (1/sqrt(320)) * exp(+2*pi*i * (m-160)*(k-160)/320)
//    Integer phase reduced mod 320 before float conversion for full accuracy.
// ---------------------------------------------------------------------------
__global__ void build_w_kernel(float* __restrict__ Wr, float* __restrict__ Wi) {
    int t = blockIdx.x * blockDim.x + threadIdx.x;
    if (t >= HW) return;
    int m = t / RES, k = t % RES;
    long long prod = (long long)(m - RES / 2) * (long long)(k - RES / 2);
    int ph = (int)(prod % RES);
    if (ph < 0) ph += RES;
    const float two_pi_over_n = 6.28318530717958647692f / (float)RES;
    float theta = two_pi_over_n * (float)ph;
    float s, c;
    sincosf(theta, &s, &c);
    const float scale = 0.055901699437494742410f; // 1/sqrt(320)
    Wr[t] = scale * c;
    Wi[t] = scale * s;
}

// ---------------------------------------------------------------------------
// 2) Zero the scatter accumulation grid (B*2*H*W floats).
// ---------------------------------------------------------------------------
__global__ void zero_kernel(float4* __restrict__ p, int n4) {
    int t = blockIdx.x * blockDim.x + threadIdx.x;
    if (t < n4) p[t] = make_float4(0.f, 0.f, 0.f, 0.f);
}

// ---------------------------------------------------------------------------
// 3) Bilinear gather of k-space at trajectory points + adjoint scatter-add.
//    One thread per (point p, batch b).  Reference semantics:
//      px = tx + W/2 - 0.5,  4 corners, out-of-range corner weight = 0.
//    sub = sum_c w_c * ksp[idx_c];  grid[idx_c] += sub * w_c
// ---------------------------------------------------------------------------
__global__ __launch_bounds__(256)
void gather_scatter_kernel(const float* __restrict__ ksp,   // [B,1,H,W,2]
                           const float* __restrict__ traj,  // [N,2]
                           float* __restrict__ grid)        // [B,2,H*W]
{
    int p = blockIdx.x * blockDim.x + threadIdx.x;
    int b = blockIdx.y;
    if (p >= NTRAJ) return;

    float tx = traj[2 * p + 0];
    float ty = traj[2 * p + 1];
    float px = tx + (0.5f * RES - 0.5f);
    float py = ty + (0.5f * RES - 0.5f);
    float x0f = floorf(px), y0f = floorf(py);
    float wx1 = px - x0f, wy1 = py - y0f;
    int x0 = (int)x0f, y0 = (int)y0f;

    int   cx[4] = { x0, x0 + 1, x0,     x0 + 1 };
    int   cy[4] = { y0, y0,     y0 + 1, y0 + 1 };
    float cw[4] = { (1.f - wx1) * (1.f - wy1), wx1 * (1.f - wy1),
                    (1.f - wx1) * wy1,         wx1 * wy1 };

    const float* kb = ksp + (size_t)b * HW * 2;

    float sub_r = 0.f, sub_i = 0.f;
    #pragma unroll
    for (int c = 0; c < 4; ++c) {
        if (cx[c] >= 0 && cx[c] < RES && cy[c] >= 0 && cy[c] < RES) {
            const float2 v = *(const float2*)(kb + ((size_t)cy[c] * RES + cx[c]) * 2);
            sub_r += cw[c] * v.x;
            sub_i += cw[c] * v.y;
        }
    }

    float* gr = grid + (size_t)(b * 2 + 0) * HW;
    float* gi = grid + (size_t)(b * 2 + 1) * HW;
    #pragma unroll
    for (int c = 0; c < 4; ++c) {
        if (cx[c] >= 0 && cx[c] < RES && cy[c] >= 0 && cy[c] < RES) {
            int idx = cy[c] * RES + cx[c];
            atomicAdd(gr + idx, sub_r * cw[c]);
            atomicAdd(gi + idx, sub_i * cw[c]);
        }
    }
}

// ---------------------------------------------------------------------------
// 4) Complex GEMM O = A * B (320x320, row-major, ld=320) via fp32 WMMA.
//    Each wave owns a 32x32 complex tile = 2x2 WMMA subtiles, K-loop step 4
//    with V_WMMA_F32_16X16X4_F32.  A fragments reused across two N-subtiles,
//    B fragments reused across two M-subtiles: 12 VMEM loads per 16 WMMAs.
//    Real part needs  Ar*Br - Ai*Bi  -> VALU-negate Ai (fp32 WMMA has no NEG).
// ---------------------------------------------------------------------------
#define WMMA_F32(A, B, C) \
    __builtin_amdgcn_wmma_f32_16x16x4_f32(false, (A), false, (B), (short)0, (C), false, false)

__global__ __launch_bounds__(128)
void zgemm_wmma_kernel(const float* __restrict__ Ar, const float* __restrict__ Ai, int strideA,
                       const float* __restrict__ Br, const float* __restrict__ Bi, int strideB,
                       float*       __restrict__ Or, float*       __restrict__ Oi, int strideO)
{
    const int LD = RES;
    int wave = threadIdx.x >> 5;
    int lane = threadIdx.x & 31;
    int half = lane >> 4;     // 0: K=0,1 / rows M=v ; 1: K=2,3 / rows M=v+8
    int lm   = lane & 15;

    int tile = blockIdx.x * 4 + wave;            // 0..99  (10x10 wave tiles)
    int b    = blockIdx.y;
    int m0   = (tile / (RES / 32)) * 32;
    int n0   = (tile % (RES / 32)) * 32;

    const float* AR = Ar + (size_t)b * strideA;
    const float* AI = Ai + (size_t)b * strideA;
    const float* BR = Br + (size_t)b * strideB;
    const float* BI = Bi + (size_t)b * strideB;

    const float* ar0 = AR + (size_t)(m0 + lm) * LD + 2 * half;
    const float* ai0 = AI + (size_t)(m0 + lm) * LD + 2 * half;
    const float* ar1 = AR + (size_t)(m0 + 16 + lm) * LD + 2 * half;
    const float* ai1 = AI + (size_t)(m0 + 16 + lm) * LD + 2 * half;
    const float* br0 = BR + (size_t)(2 * half) * LD + n0 + lm;
    const float* bi0 = BI + (size_t)(2 * half) * LD + n0 + lm;
    const float* br1 = BR + (size_t)(2 * half) * LD + n0 + 16 + lm;
    const float* bi1 = BI + (size_t)(2 * half) * LD + n0 + 16 + lm;

    v8f accr00 = {}, acci00 = {}, accr01 = {}, acci01 = {};
    v8f accr10 = {}, acci10 = {}, accr11 = {}, acci11 = {};

    for (int k = 0; k < RES; k += 4) {
        // A fragments: 16x4 fp32 -> v2f per lane (contiguous along K, 8B aligned)
        v2f fAr0 = *(const v2f*)(ar0 + k);
        v2f fAi0 = *(const v2f*)(ai0 + k);
        v2f fAr1 = *(const v2f*)(ar1 + k);
        v2f fAi1 = *(const v2f*)(ai1 + k);
        // B fragments: 4x16 fp32 -> two rows per lane, stride LD
        v2f fBr0; fBr0.x = br0[(size_t)k * LD]; fBr0.y = br0[(size_t)(k + 1) * LD];
        v2f fBi0; fBi0.x = bi0[(size_t)k * LD]; fBi0.y = bi0[(size_t)(k + 1) * LD];
        v2f fBr1; fBr1.x = br1[(size_t)k * LD]; fBr1.y = br1[(size_t)(k + 1) * LD];
        v2f fBi1; fBi1.x = bi1[(size_t)k * LD]; fBi1.y = bi1[(size_t)(k + 1) * LD];
        v2f nAi0 = -fAi0;
        v2f nAi1 = -fAi1;

        accr00 = WMMA_F32(fAr0, fBr0, accr00);
        accr00 = WMMA_F32(nAi0, fBi0, accr00);
        acci00 = WMMA_F32(fAr0, fBi0, acci00);
        acci00 = WMMA_F32(fAi0, fBr0, acci00);

        accr01 = WMMA_F32(fAr0, fBr1, accr01);
        accr01 = WMMA_F32(nAi0, fBi1, accr01);
        acci01 = WMMA_F32(fAr0, fBi1, acci01);
        acci01 = WMMA_F32(fAi0, fBr1, acci01);

        accr10 = WMMA_F32(fAr1, fBr0, accr10);
        accr10 = WMMA_F32(nAi1, fBi0, accr10);
        acci10 = WMMA_F32(fAr1, fBi0, acci10);
        acci10 = WMMA_F32(fAi1, fBr0, acci10);

        accr11 = WMMA_F32(fAr1, fBr1, accr11);
        accr11 = WMMA_F32(nAi1, fBi1, accr11);
        acci11 = WMMA_F32(fAr1, fBi1, acci11);
        acci11 = WMMA_F32(fAi1, fBr1, acci11);
    }

    float* OR = Or + (size_t)b * strideO;
    float* OI = Oi + (size_t)b * strideO;
    #pragma unroll
    for (int v = 0; v < 8; ++v) {
        int m = v + 8 * half;                   // C/D layout: VGPR v, half -> M or M+8
        OR[(size_t)(m0 + m)      * LD + n0 + lm]      = accr00[v];
        OR[(size_t)(m0 + m)      * LD + n0 + 16 + lm] = accr01[v];
        OR[(size_t)(m0 + 16 + m) * LD + n0 + lm]      = accr10[v];
        OR[(size_t)(m0 + 16 + m) * LD + n0 + 16 + lm] = accr11[v];
        OI[(size_t)(m0 + m)      * LD + n0 + lm]      = acci00[v];
        OI[(size_t)(m0 + m)      * LD + n0 + 16 + lm] = acci01[v];
        OI[(size_t)(m0 + 16 + m) * LD + n0 + lm]      = acci10[v];
        OI[(size_t)(m0 + 16 + m) * LD + n0 + 16 + lm] = acci11[v];
    }
}

// ---------------------------------------------------------------------------
// Launch: W build -> zero grid -> gather/scatter -> T = W*G -> OUT = T*W
// ---------------------------------------------------------------------------
extern "C" void kernel_launch(void* const* d_in, const int* in_sizes, int n_in,
                              void* d_out, int out_size, void* d_ws, size_t ws_size,
                              hipStream_t stream) {
    const float* ksp  = (const float*)d_in[0];   // [64,1,320,320,2]
    const float* traj = (const float*)d_in[1];   // [102400,2]
    float* out = (float*)d_out;                  // [64,1,2,320,320]

    float* wsf  = (float*)d_ws;
    float* Wr   = wsf;                            // 102400 floats
    float* Wi   = wsf + HW;                       // 102400 floats
    float* grid = wsf + 2 * HW;                   // 64*2*102400 floats
    float* T    = grid + (size_t)NB * 2 * HW;     // 64*2*102400 floats

    // 1) DFT matrix
    build_w_kernel<<<(HW + 255) / 256, 256, 0, stream>>>(Wr, Wi);

    // 2) zero scatter grid
    {
        int n4 = (NB * 2 * HW) / 4;
        zero_kernel<<<(n4 + 255) / 256, 256, 0, stream>>>((float4*)grid, n4);
    }

    // 3) gather + adjoint scatter
    {
        dim3 g((NTRAJ + 255) / 256, NB);
        gather_scatter_kernel<<<g, 256, 0, stream>>>(ksp, traj, grid);
    }

    // 4) T = W * G   (A = W shared across batch, B = grid per batch)
    {
        dim3 g((RES / 32) * (RES / 32) / 4, NB);   // 25 x 64 blocks, 4 waves each
        zgemm_wmma_kernel<<<g, 128, 0, stream>>>(
            Wr, Wi, 0,
            grid, grid + HW, 2 * HW,
            T,    T    + HW, 2 * HW);
    }

    // 5) OUT = T * W  (A = T per batch, B = W shared)
    {
        dim3 g((RES / 32) * (RES / 32) / 4, NB);
        zgemm_wmma_kernel<<<g, 128, 0, stream>>>(
            T,   T   + HW, 2 * HW,
            Wr,  Wi,       0,
            out, out + HW, 2 * HW);
    }
}